// GCN_59090160058847
// MI455X (gfx1250) — compile-verified
//
#include <hip/hip_runtime.h>

// ---------------------------------------------------------------------------
// GCN 2-layer forward on MI455X (gfx1250), fp32 end-to-end.
//   norms -> SpMM1 (edge-parallel, L2-resident atomics) ->
//   fused [A*norm_dst] @ W1 +b1, relu, *p, @ W2, *norm_src
//     (V_WMMA_F32_16X16X4_F32 + TENSOR_LOAD_TO_LDS staging)
//   -> SpMM2 -> *norm_dst + b2
// ---------------------------------------------------------------------------

#define IN_DIM  128
#define HID_DIM 256
#define OUT_DIM 64
#define ROWS_PER_WG 128   // 8 row-tiles of 16
#define MT (ROWS_PER_WG / 16)

typedef float        v2f __attribute__((ext_vector_type(2)));
typedef float        v8f __attribute__((ext_vector_type(8)));
typedef unsigned int v4u __attribute__((ext_vector_type(4)));
typedef int          v4i __attribute__((ext_vector_type(4)));
typedef int          v8i __attribute__((ext_vector_type(8)));

static __device__ __forceinline__ v8f wmma_f32_4(v2f a, v2f b, v8f c) {
    // D = A(16x4,f32) * B(4x16,f32) + C(16x16,f32)
    return __builtin_amdgcn_wmma_f32_16x16x4_f32(
        /*neg_a=*/false, a, /*neg_b=*/false, b,
        /*c_mod=*/(short)0, c, /*reuse_a=*/false, /*reuse_b=*/false);
}

// LDS generic pointer -> LDS byte address (flat aperture: LDS addr = addr[31:0])
static __device__ __forceinline__ uint32_t lds_addr32(const void* p) {
    return (uint32_t)(uintptr_t)p;
}

// Tensor Data Mover: 2-D tile (or 1-D when tile_d1==0) of 4-byte elements,
// global -> LDS.  D# packed per CDNA5 ISA §8.3/§8.4; groups 2/3 zero (<=2D).
// This toolchain exposes the 6-arg builtin:
//   (uint32x4 g0, int32x8 g1, int32x4 g2, int32x4 g3, int32x8 g4, i32 cpol)
static __device__ __forceinline__ void tdm_load_2d(uint32_t lds_byte,
                                                   const void* gsrc,
                                                   uint32_t tile_d0,    // elems per row
                                                   uint32_t tile_d1,    // rows (0 => 1D)
                                                   uint32_t stride_d0,  // tensor dim0 stride, elems
                                                   uint32_t tensor_d1)  // rows in tensor (OOB -> 0)
{
    uint64_t ga = (uint64_t)(uintptr_t)gsrc;
    v4u g0;
    g0[0] = 1u;                                                  // count=1 (valid user D#)
    g0[1] = lds_byte;                                            // lds_addr (bytes)
    g0[2] = (uint32_t)ga;                                        // global_addr[31:0]
    g0[3] = (uint32_t)((ga >> 32) & 0x01FFFFFFu) | (2u << 30);   // addr[56:32] | type=2
    v8i g1;
    g1[0] = (int)(2u << 16);                                     // data_size = 4 bytes
    g1[1] = (int)((tile_d0 & 0xFFFFu) << 16);                    // tensor_dim0[15:0]
    g1[2] = (int)(((tile_d0 >> 16) & 0xFFFFu) |
                  ((tensor_d1 & 0xFFFFu) << 16));                // tensor_dim0[31:16] | dim1[15:0]
    g1[3] = (int)(((tensor_d1 >> 16) & 0xFFFFu) |
                  ((tile_d0 & 0xFFFFu) << 16));                  // dim1[31:16] | tile_dim0
    g1[4] = (int)(tile_d1 & 0xFFFFu);                            // tile_dim1 ; tile_dim2=0
    g1[5] = (int)stride_d0;                                      // tensor_dim0_stride[31:0]
    g1[6] = 0;                                                   // stride hi | dim1_stride lo
    g1[7] = 0;
    v4i z4 = {0, 0, 0, 0};
    v8i z8 = {0, 0, 0, 0, 0, 0, 0, 0};
    __builtin_amdgcn_tensor_load_to_lds(g0, g1, z4, z4, z8, 0);
}

// ------------------------- degrees & norms ---------------------------------

__global__ __launch_bounds__(256) void k_degrees(const int* __restrict__ src,
                                                 const int* __restrict__ dst,
                                                 float* __restrict__ deg_out,
                                                 float* __restrict__ deg_in,
                                                 int n_edges) {
    int e = blockIdx.x * blockDim.x + threadIdx.x;
    if (e >= n_edges) return;
    atomicAdd(&deg_out[src[e]], 1.0f);
    atomicAdd(&deg_in[dst[e]], 1.0f);
}

__global__ __launch_bounds__(256) void k_norms(const float* __restrict__ deg_out,
                                               const float* __restrict__ deg_in,
                                               float* __restrict__ nsrc,
                                               float* __restrict__ ndst,
                                               int n_nodes) {
    int i = blockIdx.x * blockDim.x + threadIdx.x;
    if (i >= n_nodes) return;
    nsrc[i] = rsqrtf(fmaxf(deg_out[i], 1.0f));
    ndst[i] = rsqrtf(fmaxf(deg_in[i], 1.0f));
}

// ------------------------- SpMM (edge-parallel) ----------------------------
// One wave32 per edge. h rows (51 MB) and agg targets stay resident in the
// 192 MB L2, so gathers and scatter-atomics run at L2 rather than HBM rates.

__global__ __launch_bounds__(256) void k_spmm1(const float* __restrict__ h,
                                               const float* __restrict__ nsrc,
                                               const int* __restrict__ src,
                                               const int* __restrict__ dst,
                                               float* __restrict__ agg1,
                                               int n_edges) {
    int wid  = (blockIdx.x * blockDim.x + threadIdx.x) >> 5;
    int lane = threadIdx.x & 31;
    if (wid >= n_edges) return;
    int s = src[wid], d = dst[wid];
    float ns = nsrc[s];
    float4 v = ((const float4*)(h + (size_t)s * IN_DIM))[lane];   // coalesced 512B/wave
    float* out = agg1 + (size_t)d * IN_DIM + lane * 4;
    atomicAdd(out + 0, v.x * ns);
    atomicAdd(out + 1, v.y * ns);
    atomicAdd(out + 2, v.z * ns);
    atomicAdd(out + 3, v.w * ns);
}

__global__ __launch_bounds__(256) void k_spmm2(const float* __restrict__ y1,
                                               const int* __restrict__ src,
                                               const int* __restrict__ dst,
                                               float* __restrict__ agg2,
                                               int n_edges) {
    int wid  = (blockIdx.x * blockDim.x + threadIdx.x) >> 5;
    int lane = threadIdx.x & 31;
    if (wid >= n_edges) return;
    int s = src[wid], d = dst[wid];
    float2 v = ((const float2*)(y1 + (size_t)s * OUT_DIM))[lane];
    float* out = agg2 + (size_t)d * OUT_DIM + lane * 2;
    atomicAdd(out + 0, v.x);
    atomicAdd(out + 1, v.y);
}

// ------------------------- fused WMMA MLP ----------------------------------

__global__ __launch_bounds__(256) void k_fused_mlp(const float* __restrict__ agg1,
                                                   const float* __restrict__ W1,
                                                   const float* __restrict__ b1,
                                                   const float* __restrict__ W2,
                                                   const float* __restrict__ p,
                                                   const float* __restrict__ nsrc,
                                                   const float* __restrict__ ndst,
                                                   float* __restrict__ y1,
                                                   int n_nodes) {
    extern __shared__ float lds[];
    float* sW1 = lds;                       // 128*256 f32 = 128 KB
    float* sH  = sW1 + IN_DIM * HID_DIM;    // 16*256  f32 =  16 KB
    float* sA  = sH + 16 * HID_DIM;         // 2x 16*128   =  16 KB  (160 KB -> 2 WG/WGP)

    const int tid     = threadIdx.x;
    const int wave    = tid >> 5;
    const int lane    = tid & 31;
    const int mn      = lane & 15;          // row (A) / col (B) within 16
    const int kb      = (lane >> 4) << 1;   // K sub-offset per ISA 16x4 layout
    const int rowBase = blockIdx.x * ROWS_PER_WG;

    // --- TDM staging: W1 (1-D, 128 KB) and the first A tile (16x128) -------
    if (wave == 0) {
        tdm_load_2d(lds_addr32(sW1), W1, (uint32_t)(IN_DIM * HID_DIM), 0u,
                    (uint32_t)(IN_DIM * HID_DIM), 1u);
    } else if (wave == 1) {
        uint32_t trows = (rowBase < n_nodes) ? (uint32_t)(n_nodes - rowBase) : 0u;
        tdm_load_2d(lds_addr32(sA), agg1 + (size_t)rowBase * IN_DIM,
                    IN_DIM, 16u, IN_DIM, trows);
    }
    __builtin_amdgcn_s_wait_tensorcnt(0);   // no-op for waves with TENSORcnt==0
    __syncthreads();

    for (int rt = 0; rt < MT; ++rt) {
        const float* curA = sA + (rt & 1) * (16 * IN_DIM);
        float*       nxtA = sA + ((rt + 1) & 1) * (16 * IN_DIM);
        const int row0 = rowBase + rt * 16;

        // Wave 7 kicks off the async TDM for the next A tile; it completes
        // in the background while all waves run GEMM1 (+GEMM2).
        if (wave == 7 && rt + 1 < MT) {
            const int r0n = rowBase + (rt + 1) * 16;
            uint32_t trows = (r0n < n_nodes) ? (uint32_t)(n_nodes - r0n) : 0u;
            tdm_load_2d(lds_addr32(nxtA), agg1 + (size_t)r0n * IN_DIM,
                        IN_DIM, 16u, IN_DIM, trows);
        }

        // ---- GEMM1: 16x128 @ 128x256 ; wave owns 2 adjacent N-tiles,
        //      single A fragment shared by both accumulators -----------------
        const int rowm = row0 + mn;
        const float ndm = (rowm < n_nodes) ? ndst[rowm] : 0.f;  // row scale (also zeroes tail)
        const int ncol0 = wave * 32;
        v8f acc0 = {0.f, 0.f, 0.f, 0.f, 0.f, 0.f, 0.f, 0.f};
        v8f acc1 = {0.f, 0.f, 0.f, 0.f, 0.f, 0.f, 0.f, 0.f};
        #pragma unroll 8
        for (int ks = 0; ks < IN_DIM / 4; ++ks) {
            const int k0 = ks * 4;
            v2f a, b0, b1v;
            a.x = curA[mn * IN_DIM + k0 + kb] * ndm;
            a.y = curA[mn * IN_DIM + k0 + kb + 1] * ndm;
            b0.x  = sW1[(k0 + kb) * HID_DIM + ncol0 + mn];
            b0.y  = sW1[(k0 + kb + 1) * HID_DIM + ncol0 + mn];
            b1v.x = sW1[(k0 + kb) * HID_DIM + ncol0 + 16 + mn];
            b1v.y = sW1[(k0 + kb + 1) * HID_DIM + ncol0 + 16 + mn];
            acc0 = wmma_f32_4(a, b0, acc0);
            acc1 = wmma_f32_4(a, b1v, acc1);
        }
        // epilogue: +b1, relu, * clamp(p,0,1) -> hidden tile in LDS
        {
            const int c0 = ncol0 + mn, c1 = ncol0 + 16 + mn;
            const float bias0 = b1[c0], bias1 = b1[c1];
            const float pr0 = fminf(fmaxf(p[c0], 0.f), 1.f);
            const float pr1 = fminf(fmaxf(p[c1], 0.f), 1.f);
            #pragma unroll
            for (int r = 0; r < 8; ++r) {
                const int ml = r + ((lane >> 4) << 3);   // C-matrix VGPR layout
                float v0 = acc0[r] + bias0; v0 = v0 > 0.f ? v0 : 0.f;
                float v1 = acc1[r] + bias1; v1 = v1 > 0.f ? v1 : 0.f;
                sH[ml * HID_DIM + c0] = v0 * pr0;
                sH[ml * HID_DIM + c1] = v1 * pr1;
            }
        }
        __syncthreads();

        if (wave < 4) {
            // ---- GEMM2: 16x256 @ 256x64 ; waves 0..3 own the 4 N-tiles ----
            const int nc0 = wave * 16;
            v8f acc = {0.f, 0.f, 0.f, 0.f, 0.f, 0.f, 0.f, 0.f};
            #pragma unroll 8
            for (int ks = 0; ks < HID_DIM / 4; ++ks) {
                const int k0 = ks * 4;
                v2f a, b;
                a.x = sH[mn * HID_DIM + k0 + kb];
                a.y = sH[mn * HID_DIM + k0 + kb + 1];
                b.x = W2[(k0 + kb) * OUT_DIM + nc0 + mn];      // 64 KB, L2-hot
                b.y = W2[(k0 + kb + 1) * OUT_DIM + nc0 + mn];
                acc = wmma_f32_4(a, b, acc);
            }
            const int ncol = nc0 + mn;
            #pragma unroll
            for (int r = 0; r < 8; ++r) {
                const int ml = r + ((lane >> 4) << 3);
                const int row = row0 + ml;
                if (row < n_nodes)
                    y1[(size_t)row * OUT_DIM + ncol] = acc[r] * nsrc[row];
            }
        }
        // TDM producer drains its tensor counter before consumers cross the barrier.
        if (wave == 7)
            __builtin_amdgcn_s_wait_tensorcnt(0);
        __syncthreads();
    }
}

// ------------------------- epilogue ----------------------------------------

__global__ __launch_bounds__(256) void k_epilogue(const float* __restrict__ agg2,
                                                  const float* __restrict__ ndst,
                                                  const float* __restrict__ b2,
                                                  float* __restrict__ out,
                                                  int n_nodes) {
    long long idx = (long long)blockIdx.x * blockDim.x + threadIdx.x;
    long long total = (long long)n_nodes * OUT_DIM;
    if (idx >= total) return;
    int row = (int)(idx >> 6);
    int col = (int)(idx & 63);
    out[idx] = agg2[idx] * ndst[row] + b2[col];
}

// ------------------------- launcher ----------------------------------------

extern "C" void kernel_launch(void* const* d_in, const int* in_sizes, int n_in,
                              void* d_out, int out_size, void* d_ws, size_t ws_size,
                              hipStream_t stream) {
    const float* h   = (const float*)d_in[0];
    const float* W1  = (const float*)d_in[1];
    const float* b1  = (const float*)d_in[2];
    const float* W2  = (const float*)d_in[3];
    const float* b2  = (const float*)d_in[4];
    const float* p   = (const float*)d_in[5];
    const int*   src = (const int*)d_in[6];
    const int*   dst = (const int*)d_in[7];
    // n_nodes lives on device as a 1-elem array; derive sizes from in_sizes.
    const int n_nodes = in_sizes[0] / IN_DIM;
    const int n_edges = in_sizes[6];
    float* out = (float*)d_out;

    // Workspace layout (floats)
    float* ws       = (float*)d_ws;
    float* deg_out  = ws;
    float* deg_in   = deg_out + n_nodes;
    float* nsrc     = deg_in + n_nodes;
    float* ndst     = nsrc + n_nodes;
    float* agg1     = ndst + n_nodes;                       // n*128
    float* y1       = agg1 + (size_t)n_nodes * IN_DIM;      // n*64
    float* agg2     = y1 + (size_t)n_nodes * OUT_DIM;       // n*64

    // Zero atomic accumulators (graph-capturable memsets).
    (void)hipMemsetAsync(deg_out, 0, sizeof(float) * 2 * n_nodes, stream);
    (void)hipMemsetAsync(agg1, 0, sizeof(float) * (size_t)n_nodes * IN_DIM, stream);
    (void)hipMemsetAsync(agg2, 0, sizeof(float) * (size_t)n_nodes * OUT_DIM, stream);

    k_degrees<<<(n_edges + 255) / 256, 256, 0, stream>>>(src, dst, deg_out, deg_in, n_edges);
    k_norms<<<(n_nodes + 255) / 256, 256, 0, stream>>>(deg_out, deg_in, nsrc, ndst, n_nodes);

    // SpMM1: one wave per edge (8 edges / 256-thread block)
    k_spmm1<<<(n_edges + 7) / 8, 256, 0, stream>>>(h, nsrc, src, dst, agg1, n_edges);

    // Fused MLP: 160 KB dynamic LDS (CDNA5: up to 320 KB/WG)
    const size_t ldsBytes =
        sizeof(float) * (IN_DIM * HID_DIM + 16 * HID_DIM + 2 * 16 * IN_DIM);
    k_fused_mlp<<<(n_nodes + ROWS_PER_WG - 1) / ROWS_PER_WG, 256, ldsBytes, stream>>>(
        agg1, W1, b1, W2, p, nsrc, ndst, y1, n_nodes);

    // SpMM2: one wave per edge
    k_spmm2<<<(n_edges + 7) / 8, 256, 0, stream>>>(y1, src, dst, agg2, n_edges);

    const long long total = (long long)n_nodes * OUT_DIM;
    k_epilogue<<<(int)((total + 255) / 256), 256, 0, stream>>>(agg2, ndst, b2, out, n_nodes);
}